// MultiViewDGT_44916767981861
// MI455X (gfx1250) — compile-verified
//
#include <hip/hip_runtime.h>
#include <hip/hip_bf16.h>
#include <math.h>

// ---------------- problem constants (from reference) ----------------
enum { N_NODES = 50000, EV = 200000, XD = 64, HID = 128, NH = 4, HD = 128,
       G_GRP = 2000, P_PORT = 100000, B_BATCH = 4096, MKT = 16, QKV = NH * HD };

typedef float v2f __attribute__((ext_vector_type(2)));
typedef float v8f __attribute__((ext_vector_type(8)));

static inline int cdiv(int a, int b) { return (a + b - 1) / b; }

__device__ inline float sigf(float x) { return 1.f / (1.f + expf(-x)); }
__device__ inline unsigned f2ord(float f) {
  unsigned u = __float_as_uint(f);
  return (u & 0x80000000u) ? ~u : (u | 0x80000000u);
}
__device__ inline float ord2f(unsigned u) {
  return (u & 0x80000000u) ? __uint_as_float(u & 0x7fffffffu) : __uint_as_float(~u);
}

// ---------------- generic WMMA fp32 GEMM: C = act(A[n,k] @ B[k,m] + bias) ----
// block = 256 threads = 8 waves; block tile 64x32; wave tile 16x16; K tile 32.
#define KT 32
__global__ __launch_bounds__(256)
void k_gemm(const float* __restrict__ A, const float* __restrict__ Bm,
            const float* __restrict__ bias, float* __restrict__ C,
            int n, int k, int m, int act)
{
  __shared__ float As[64][KT + 1];
  __shared__ float Bs[KT][32 + 1];
  const int tid  = threadIdx.x;
  const int wave = tid >> 5, lane = tid & 31;
  const int half = lane >> 4, lm = lane & 15;
  const int wr = wave >> 1, wc = wave & 1;
  const int rowBase = blockIdx.x * 64;
  const int colBase = blockIdx.y * 32;

  v8f acc = {};
  for (int k0 = 0; k0 < k; k0 += KT) {
    for (int i = tid; i < 64 * KT; i += 256) {
      int rr = i >> 5, cc = i & (KT - 1);
      int gr = rowBase + rr, gc = k0 + cc;
      As[rr][cc] = (gr < n && gc < k) ? A[(size_t)gr * k + gc] : 0.f;
    }
    for (int i = tid; i < KT * 32; i += 256) {
      int rr = i >> 5, cc = i & 31;
      int gr = k0 + rr, gc = colBase + cc;
      Bs[rr][cc] = (gr < k && gc < m) ? Bm[(size_t)gr * m + gc] : 0.f;
    }
    __syncthreads();
#pragma unroll
    for (int kk = 0; kk < KT; kk += 4) {
      v2f a, b;
      int ks = kk + 2 * half;
      a.x = As[wr * 16 + lm][ks];
      a.y = As[wr * 16 + lm][ks + 1];
      b.x = Bs[ks][wc * 16 + lm];
      b.y = Bs[ks + 1][wc * 16 + lm];
      acc = __builtin_amdgcn_wmma_f32_16x16x4_f32(false, a, false, b,
                                                  (short)0, acc, false, false);
    }
    __syncthreads();
  }
  int gc = colBase + wc * 16 + lm;
  if (gc < m) {
    float bv = bias ? bias[gc] : 0.f;
#pragma unroll
    for (int r8 = 0; r8 < 8; ++r8) {
      int gr = rowBase + wr * 16 + r8 + 8 * half;
      if (gr < n) {
        float v = acc[r8] + bv;
        if (act) v = fmaxf(v, 0.f);
        C[(size_t)gr * m + gc] = v;
      }
    }
  }
}

// ---------------- small utility kernels ----------------
__global__ void k_fill_f32(float* p, float v, int n) {
  int t = blockIdx.x * blockDim.x + threadIdx.x; if (t < n) p[t] = v;
}
__global__ void k_fill_u32(unsigned* p, unsigned v, int n) {
  int t = blockIdx.x * blockDim.x + threadIdx.x; if (t < n) p[t] = v;
}
__global__ void k_copy(const float* s, float* d, int n) {
  int t = blockIdx.x * blockDim.x + threadIdx.x; if (t < n) d[t] = s[t];
}

// scalars: [0..7] view coefs per layer, [8]=cg, [9]=pfg, [16..21] ew sum/sumsq
__global__ void k_scalars(const float* gl, const float* cg, const float* pg, float* s) {
  if (threadIdx.x == 0 && blockIdx.x == 0) {
    for (int l = 0; l < 2; ++l) {
      float g0 = sigf(gl[l*4+0]), g1 = sigf(gl[l*4+1]),
            g2 = sigf(gl[l*4+2]), g3 = sigf(gl[l*4+3]);
      s[l*4+0] = g0 - g1 - g2 - g3; s[l*4+1] = g1; s[l*4+2] = g2; s[l*4+3] = g3;
    }
    s[8] = sigf(cg[0]); s[9] = sigf(pg[0]);
    for (int i = 16; i < 22; ++i) s[i] = 0.f;
  }
}

__global__ __launch_bounds__(256)
void k_ew_stats(const float* __restrict__ ewv, float* sums, int E) {
  __shared__ float sh0[256], sh1[256];
  int v = blockIdx.y;                 // views 1..3 -> v+1
  float s = 0.f, ss = 0.f;
  for (int i = blockIdx.x * blockDim.x + threadIdx.x; i < E; i += gridDim.x * blockDim.x) {
    float x = ewv[(size_t)(v + 1) * E + i]; s += x; ss += x * x;
  }
  sh0[threadIdx.x] = s; sh1[threadIdx.x] = ss; __syncthreads();
  for (int o = 128; o > 0; o >>= 1) {
    if (threadIdx.x < o) { sh0[threadIdx.x] += sh0[threadIdx.x+o]; sh1[threadIdx.x] += sh1[threadIdx.x+o]; }
    __syncthreads();
  }
  if (threadIdx.x == 0) { atomicAdd(&sums[16 + v*2], sh0[0]); atomicAdd(&sums[16 + v*2 + 1], sh1[0]); }
}

__global__ void k_ew_norm(const float* __restrict__ ewv, const float* __restrict__ scal,
                          float* __restrict__ ewn, int E) {
  int t = blockIdx.x * blockDim.x + threadIdx.x;
  if (t >= 3 * E) return;
  int v = t / E, e = t - v * E;
  float mu = scal[16 + v*2] / (float)E;
  float var = scal[16 + v*2 + 1] / (float)E - mu * mu;
  float sd = sqrtf(fmaxf(var, 0.f)) + 1e-6f;
  float scale = (v == 0) ? 1.f : scal[8];       // views 2,3 multiply by cg
  ewn[t] = (ewv[(size_t)(v + 1) * E + e] - mu) / sd * scale;
}

// layernorm over HID=128, wave per row (wave32)
__global__ __launch_bounds__(256)
void k_layernorm(const float* __restrict__ in, const float* __restrict__ g,
                 const float* __restrict__ b, float* __restrict__ out, int n) {
  int row = (blockIdx.x * blockDim.x + threadIdx.x) >> 5;
  int lane = threadIdx.x & 31;
  if (row >= n) return;
  const float* r = in + (size_t)row * HID;
  float v[4]; float s = 0.f;
#pragma unroll
  for (int i = 0; i < 4; ++i) { v[i] = r[lane*4+i]; s += v[i]; }
#pragma unroll
  for (int o = 16; o > 0; o >>= 1) s += __shfl_xor(s, o, 32);
  float mu = s * (1.f / HID);
  float q = 0.f;
#pragma unroll
  for (int i = 0; i < 4; ++i) { float d = v[i]-mu; q += d*d; }
#pragma unroll
  for (int o = 16; o > 0; o >>= 1) q += __shfl_xor(q, o, 32);
  float rstd = rsqrtf(q * (1.f / HID) + 1e-5f);
  float* w = out + (size_t)row * HID;
#pragma unroll
  for (int i = 0; i < 4; ++i) {
    int d = lane*4+i;
    w[d] = (v[i]-mu)*rstd*g[d] + b[d];
  }
}

// ---------------- edge kernels ----------------
__global__ __launch_bounds__(256)
void k_edge_score(const float* __restrict__ q, const float* __restrict__ k,
                  const float* __restrict__ We, const int* __restrict__ src,
                  const int* __restrict__ dst, const float* __restrict__ ew,
                  float* __restrict__ score, unsigned* __restrict__ mmax, int E) {
  int t = blockIdx.x * blockDim.x + threadIdx.x;
  if (t >= E * NH) return;
  int e = t >> 2, h = t & 3;
  int s = src[e], d = dst[e];
  float w = ew[e];
  const float* qr = q + (size_t)d * QKV + h * HD;
  const float* kr = k + (size_t)s * QKV + h * HD;
  const float* wr = We + h * HD;
  float acc = 0.f;
#pragma unroll 8
  for (int i = 0; i < HD; ++i) acc += qr[i] * (kr[i] + w * wr[i]);
  acc *= 0.08838834764831845f;          // 1/sqrt(HD)
  score[t] = acc;
  atomicMax(&mmax[(size_t)d * NH + h], f2ord(acc));
}

__global__ __launch_bounds__(256)
void k_edge_exp(float* __restrict__ score, const unsigned* __restrict__ mmax,
                float* __restrict__ den, const int* __restrict__ dst, int E) {
  int t = blockIdx.x * blockDim.x + threadIdx.x;
  if (t >= E * NH) return;
  int e = t >> 2, h = t & 3;
  int d = dst[e];
  float ex = expf(score[t] - ord2f(mmax[(size_t)d * NH + h]));
  score[t] = ex;
  atomicAdd(&den[(size_t)d * NH + h], ex);
}

__global__ __launch_bounds__(256)
void k_edge_msg(const float* __restrict__ v, const float* __restrict__ We,
                const int* __restrict__ src, const int* __restrict__ dst,
                const float* __restrict__ ew, const float* __restrict__ ex,
                const float* __restrict__ den, float* __restrict__ out, int E) {
  int t = blockIdx.x * blockDim.x + threadIdx.x;
  if (t >= E * HD) return;
  int e = t >> 7, dch = t & 127;
  int s = src[e], d = dst[e];
  float w = ew[e];
  float acc = 0.f;
#pragma unroll
  for (int h = 0; h < NH; ++h) {
    float alpha = ex[(size_t)e * NH + h] / den[(size_t)d * NH + h];
    acc += alpha * (v[(size_t)s * QKV + h * HD + dch] + w * We[h * HD + dch]);
  }
  atomicAdd(&out[(size_t)d * HID + dch], acc * (1.f / NH));
}

// beta gate + accumulate coef * tview into hacc
__global__ __launch_bounds__(256)
void k_beta_acc(const float* __restrict__ out, const float* __restrict__ r,
                const float* __restrict__ Wb, const float* __restrict__ scal,
                int coefIdx, float* __restrict__ hacc, int n) {
  int i = blockIdx.x * blockDim.x + threadIdx.x;
  if (i >= n) return;
  const float* o = out + (size_t)i * HID;
  const float* rr = r + (size_t)i * HID;
  float s = 0.f;
#pragma unroll 8
  for (int d = 0; d < HID; ++d) {
    float od = o[d], rd = rr[d];
    s += od * Wb[d] + rd * Wb[HID + d] + (od - rd) * Wb[2 * HID + d];
  }
  float beta = sigf(s);
  float coef = scal[coefIdx];
  float* h = hacc + (size_t)i * HID;
#pragma unroll 8
  for (int d = 0; d < HID; ++d)
    h[d] += coef * (beta * rr[d] + (1.f - beta) * o[d]);
}

// ---------------- portfolio kernels ----------------
__global__ __launch_bounds__(256)
void k_port_acc(const float* __restrict__ h2, const int* __restrict__ pn,
                const int* __restrict__ pg, const float* __restrict__ wsgn,
                float* __restrict__ Pa, float* __restrict__ Ps, int P) {
  int t = blockIdx.x * blockDim.x + threadIdx.x;
  if (t >= P * HID) return;
  int p = t >> 7, d = t & 127;
  int node = pn[p], g = pg[p];
  float w = wsgn[p], wa = fabsf(w);
  float hv = h2[(size_t)node * HID + d];
  atomicAdd(&Pa[(size_t)g * HID + d], wa * hv);
  atomicAdd(&Ps[(size_t)g * HID + d], w * hv);
}

__global__ __launch_bounds__(256)
void k_port_loo(const float* __restrict__ h2, const int* __restrict__ pn,
                const int* __restrict__ pg, const float* __restrict__ wsgn,
                const float* __restrict__ Pa, const float* __restrict__ Ps,
                float* __restrict__ VaVs, float* __restrict__ denN, int P) {
  int t = blockIdx.x * blockDim.x + threadIdx.x;
  if (t >= P * HID) return;
  int p = t >> 7, d = t & 127;
  int node = pn[p], g = pg[p];
  float w = wsgn[p], wa = fabsf(w);
  float hv = h2[(size_t)node * HID + d];
  float looa = Pa[(size_t)g * HID + d] - wa * hv;
  float loos = Ps[(size_t)g * HID + d] - w * hv;
  atomicAdd(&VaVs[(size_t)node * 2 * HID + d], wa * looa);
  atomicAdd(&VaVs[(size_t)node * 2 * HID + HID + d], wa * loos);
  if (d == 0) atomicAdd(&denN[node], wa);
}

__global__ __launch_bounds__(256)
void k_port_norm(float* __restrict__ VaVs, const float* __restrict__ denN, int n) {
  int row = (blockIdx.x * blockDim.x + threadIdx.x) >> 5;
  int lane = threadIdx.x & 31;
  if (row >= n) return;
  float den = fmaxf(denN[row], 1e-8f);
  float* p = VaVs + (size_t)row * 2 * HID;
  float a[4], b[4]; float sa = 0.f, sb = 0.f;
#pragma unroll
  for (int i = 0; i < 4; ++i) {
    a[i] = p[lane*4+i] / den;        sa += a[i]*a[i];
    b[i] = p[HID+lane*4+i] / den;    sb += b[i]*b[i];
  }
#pragma unroll
  for (int o = 16; o > 0; o >>= 1) { sa += __shfl_xor(sa, o, 32); sb += __shfl_xor(sb, o, 32); }
  float na = fmaxf(sqrtf(sa), 1e-6f), nb = fmaxf(sqrtf(sb), 1e-6f);
#pragma unroll
  for (int i = 0; i < 4; ++i) { p[lane*4+i] = a[i]/na; p[HID+lane*4+i] = b[i]/nb; }
}

__global__ void k_add_scaled(const float* __restrict__ h2, const float* __restrict__ pf,
                             const float* __restrict__ scal, float* __restrict__ out, int n) {
  int t = blockIdx.x * blockDim.x + threadIdx.x;
  if (t < n) out[t] = h2[t] + scal[9] * pf[t];
}

// ---------------- head kernels ----------------
__global__ void k_gather_z(const float* __restrict__ h, const int* __restrict__ anchor,
                           const float* __restrict__ m2, float* __restrict__ zcat, int B) {
  int t = blockIdx.x * blockDim.x + threadIdx.x;
  if (t >= B * HID) return;
  int b = t >> 7, d = t & 127;
  zcat[(size_t)b * 2 * HID + d] = h[(size_t)anchor[b] * HID + d];
  zcat[(size_t)b * 2 * HID + HID + d] = m2[t];
}

__global__ __launch_bounds__(256)
void k_head2(const float* __restrict__ th, const float* __restrict__ W2,
             const float* __restrict__ b2, float* __restrict__ out, int B) {
  int row = (blockIdx.x * blockDim.x + threadIdx.x) >> 5;
  int lane = threadIdx.x & 31;
  if (row >= B) return;
  float s = 0.f;
#pragma unroll
  for (int i = 0; i < 4; ++i) s += th[(size_t)row * HID + lane*4+i] * W2[lane*4+i];
#pragma unroll
  for (int o = 16; o > 0; o >>= 1) s += __shfl_xor(s, o, 32);
  if (lane == 0) out[row] = s + b2[0];
}

// ---------------- host orchestration ----------------
static void gemm(const float* A, const float* B, const float* bias, float* C,
                 int n, int k, int m, int act, hipStream_t st) {
  dim3 g(cdiv(n, 64), cdiv(m, 32));
  k_gemm<<<g, 256, 0, st>>>(A, B, bias, C, n, k, m, act);
}
static void fillf(float* p, float v, int n, hipStream_t st) {
  k_fill_f32<<<cdiv(n, 256), 256, 0, st>>>(p, v, n);
}

extern "C" void kernel_launch(void* const* d_in, const int* in_sizes, int n_in,
                              void* d_out, int out_size, void* d_ws, size_t ws_size,
                              hipStream_t stream) {
  const float* x        = (const float*)d_in[0];
  const int*   anchor   = (const int*)  d_in[1];
  const float* market   = (const float*)d_in[2];
  const int*   ei       = (const int*)  d_in[3];
  const float* ewv      = (const float*)d_in[4];
  const int*   pnodes   = (const int*)  d_in[5];
  const int*   pgid     = (const int*)  d_in[6];
  const float* pwsgn    = (const float*)d_in[7];
  const float* encW1    = (const float*)d_in[8];
  const float* encb1    = (const float*)d_in[9];
  const float* encW2    = (const float*)d_in[10];
  const float* encb2    = (const float*)d_in[11];
  const float* lnG[3]   = {(const float*)d_in[12], (const float*)d_in[14], (const float*)d_in[16]};
  const float* lnB[3]   = {(const float*)d_in[13], (const float*)d_in[15], (const float*)d_in[17]};
  const float* Wq       = (const float*)d_in[18];
  const float* bq       = (const float*)d_in[19];
  const float* Wk       = (const float*)d_in[20];
  const float* bk       = (const float*)d_in[21];
  const float* Wv       = (const float*)d_in[22];
  const float* bv       = (const float*)d_in[23];
  const float* We       = (const float*)d_in[24];
  const float* Wskip    = (const float*)d_in[25];
  const float* bskip    = (const float*)d_in[26];
  const float* Wbeta    = (const float*)d_in[27];
  const float* g_logits = (const float*)d_in[28];
  const float* corrg    = (const float*)d_in[29];
  const float* pfg      = (const float*)d_in[30];
  const float* pfW      = (const float*)d_in[31];
  const float* pfb      = (const float*)d_in[32];
  const float* mktW1    = (const float*)d_in[33];
  const float* mktb1    = (const float*)d_in[34];
  const float* mktW2    = (const float*)d_in[35];
  const float* mktb2    = (const float*)d_in[36];
  const float* headW1   = (const float*)d_in[37];
  const float* headb1   = (const float*)d_in[38];
  const float* headW2   = (const float*)d_in[39];
  const float* headb2   = (const float*)d_in[40];

  float* ws = (float*)d_ws;
  size_t o = 0;
  float* SCAL = ws + o; o += 64;
  float* EWN  = ws + o; o += (size_t)3 * EV;
  float* HA   = ws + o; o += (size_t)N_NODES * HID;
  float* HB   = ws + o; o += (size_t)N_NODES * HID;
  float* QB   = ws + o; o += (size_t)N_NODES * QKV;   // q, then v; also VaVs / t1
  float* KB   = ws + o; o += (size_t)N_NODES * QKV;   // k; also Pa/Ps/denN
  float* RB   = ws + o; o += (size_t)N_NODES * HID;   // skip; also head buffers
  float* OB   = ws + o; o += (size_t)N_NODES * HID;   // out; also pf
  float* SC   = ws + o; o += (size_t)EV * NH;         // score / exp
  unsigned* MM = (unsigned*)(ws + o); o += (size_t)N_NODES * NH;
  float* DN   = ws + o; o += (size_t)N_NODES * NH;
  (void)ws_size; (void)n_in; (void)in_sizes; (void)out_size;

  // scalars + edge-weight normalization
  k_scalars<<<1, 1, 0, stream>>>(g_logits, corrg, pfg, SCAL);
  k_ew_stats<<<dim3(256, 3), 256, 0, stream>>>(ewv, SCAL, EV);
  k_ew_norm<<<cdiv(3 * EV, 256), 256, 0, stream>>>(ewv, SCAL, EWN, EV);

  // encoder: h0 = ln(relu(x@W1+b1)@W2+b2)
  gemm(x, encW1, encb1, QB, N_NODES, XD, HID, 1, stream);
  gemm(QB, encW2, encb2, RB, N_NODES, HID, HID, 0, stream);
  k_layernorm<<<cdiv(N_NODES * 32, 256), 256, 0, stream>>>(RB, lnG[0], lnB[0], HA, N_NODES);

  // two transformer layers
  for (int l = 0; l < 2; ++l) {
    const float* hin = (l == 0) ? HA : HB;
    float* hacc      = (l == 0) ? HB : HA;
    k_copy<<<cdiv(N_NODES * HID, 256), 256, 0, stream>>>(hin, hacc, N_NODES * HID);
    for (int vi = 0; vi < 4; ++vi) {
      int lv = l * 4 + vi;
      const int* src = ei + (size_t)vi * 2 * EV;
      const int* dst = src + EV;
      const float* ewp = (vi == 0) ? ewv : (EWN + (size_t)(vi - 1) * EV);
      const float* Wev = We + (size_t)lv * QKV;

      gemm(hin, Wq + (size_t)lv * HID * QKV, bq + (size_t)lv * QKV, QB, N_NODES, HID, QKV, 0, stream);
      gemm(hin, Wk + (size_t)lv * HID * QKV, bk + (size_t)lv * QKV, KB, N_NODES, HID, QKV, 0, stream);
      k_fill_u32<<<cdiv(N_NODES * NH, 256), 256, 0, stream>>>(MM, 0x007FFFFFu, N_NODES * NH);
      k_edge_score<<<cdiv(EV * NH, 256), 256, 0, stream>>>(QB, KB, Wev, src, dst, ewp, SC, MM, EV);
      fillf(DN, 0.f, N_NODES * NH, stream);
      k_edge_exp<<<cdiv(EV * NH, 256), 256, 0, stream>>>(SC, MM, DN, dst, EV);
      // v reuses q buffer (q dead after scores)
      gemm(hin, Wv + (size_t)lv * HID * QKV, bv + (size_t)lv * QKV, QB, N_NODES, HID, QKV, 0, stream);
      fillf(OB, 0.f, N_NODES * HID, stream);
      k_edge_msg<<<cdiv(EV * HD, 256), 256, 0, stream>>>(QB, Wev, src, dst, ewp, SC, DN, OB, EV);
      gemm(hin, Wskip + (size_t)lv * HID * HID, bskip + (size_t)lv * HID, RB, N_NODES, HID, HID, 0, stream);
      k_beta_acc<<<cdiv(N_NODES, 256), 256, 0, stream>>>(OB, RB, Wbeta + (size_t)lv * 3 * HID,
                                                         SCAL, lv, hacc, N_NODES);
    }
    k_layernorm<<<cdiv(N_NODES * 32, 256), 256, 0, stream>>>(hacc, lnG[l + 1], lnB[l + 1], hacc, N_NODES);
  }
  float* H2 = HA;   // h2 lives in HA after layer 1

  // portfolio leave-one-out
  float* Pa   = KB;
  float* Ps   = KB + (size_t)G_GRP * HID;
  float* denN = KB + (size_t)2 * G_GRP * HID;
  float* VaVs = QB;                      // (N, 256)
  fillf(Pa, 0.f, 2 * G_GRP * HID, stream);
  fillf(denN, 0.f, N_NODES, stream);
  k_port_acc<<<cdiv(P_PORT * HID, 256), 256, 0, stream>>>(H2, pnodes, pgid, pwsgn, Pa, Ps, P_PORT);
  fillf(VaVs, 0.f, N_NODES * 2 * HID, stream);
  k_port_loo<<<cdiv(P_PORT * HID, 256), 256, 0, stream>>>(H2, pnodes, pgid, pwsgn, Pa, Ps, VaVs, denN, P_PORT);
  k_port_norm<<<cdiv(N_NODES * 32, 256), 256, 0, stream>>>(VaVs, denN, N_NODES);
  gemm(VaVs, pfW, pfb, OB, N_NODES, 2 * HID, HID, 0, stream);
  k_add_scaled<<<cdiv(N_NODES * HID, 256), 256, 0, stream>>>(H2, OB, SCAL, HB, N_NODES * HID);

  // head
  float* mt   = RB;
  float* m2   = RB + (size_t)B_BATCH * HID;
  float* zcat = RB + (size_t)2 * B_BATCH * HID;
  float* th   = RB + (size_t)4 * B_BATCH * HID;
  gemm(market, mktW1, mktb1, mt, B_BATCH, MKT, HID, 1, stream);
  gemm(mt, mktW2, mktb2, m2, B_BATCH, HID, HID, 0, stream);
  k_gather_z<<<cdiv(B_BATCH * HID, 256), 256, 0, stream>>>(HB, anchor, m2, zcat, B_BATCH);
  gemm(zcat, headW1, headb1, th, B_BATCH, 2 * HID, HID, 1, stream);
  k_head2<<<cdiv(B_BATCH * 32, 256), 256, 0, stream>>>(th, headW2, headb2, (float*)d_out, B_BATCH);
}